// ExpertsPreferredMaskedRouter_19095424598757
// MI455X (gfx1250) — compile-verified
//
#include <hip/hip_runtime.h>

#define NG 4
#define NT 2048
#define ND 1024
#define NE 8
#define NGT (NG * NT)
#define NBLK1 (NGT / 128)   // 64 blocks for the logits kernel
#define WSTRIDE 1028        // LDS row stride in floats: 1028 % 64 == 4 ->
                            // lane n (n=0..15) hits banks 4n..4n+3 on b128
                            // reads: every bank hit exactly twice per wave.

typedef __attribute__((ext_vector_type(2))) float v2f;
typedef __attribute__((ext_vector_type(4))) float v4f;
typedef __attribute__((ext_vector_type(8))) float v8f;

// ---------------------------------------------------------------------------
// Zero the whole output (dispatch + combine + 3 losses). Poison is 0xAA and is
// not re-cleared between replays, so every call must re-zero.
// ---------------------------------------------------------------------------
__global__ void zero_kernel(int* __restrict__ p, long long n) {
    long long i = (long long)blockIdx.x * blockDim.x + threadIdx.x;
    long long stride = (long long)gridDim.x * blockDim.x;
    long long n4 = n >> 2;
    int4* p4 = (int4*)p;
    for (long long j = i; j < n4; j += stride) p4[j] = make_int4(0, 0, 0, 0);
    for (long long j = (n4 << 2) + i; j < n; j += stride) p[j] = 0;
}

// ---------------------------------------------------------------------------
// Logits via V_WMMA_F32_16X16X4_F32 + softmax + z-loss partials.
// 256 threads = 8 waves; wave w computes a 16-token x 16-col (8 experts used)
// tile. K loop runs in steps of 8 columns = 2 WMMAs fed by one b128 load pair.
// Column mapping for the step pair starting at k (free choice, A/B agree):
//   WMMA 0: cols { k+0,k+1 (lanes kh=0) , k+4,k+5 (lanes kh=1) }
//   WMMA 1: cols { k+2,k+3 (lanes kh=0) , k+6,k+7 (lanes kh=1) }
// so each lane loads float4 at col k + 4*kh and feeds lo/hi halves.
// W is staged in LDS padded to 16 rows (rows 8..15 zero) -> branchless B
// loads, EXEC stays all-1s through the whole loop (WMMA requirement, and no
// per-iteration s_and_saveexec).
// ---------------------------------------------------------------------------
__global__ __launch_bounds__(256) void logits_kernel(
    const float* __restrict__ x, const float* __restrict__ W,
    const float* __restrict__ bias, float* __restrict__ probs,
    float* __restrict__ zpartial)
{
    __shared__ float wlds[16][WSTRIDE];  // 16 rows x 1028 floats = 65.8 KiB
    __shared__ float lg[128][NE + 1];    // +1 pad: conflict-free softmax reads
    __shared__ float zsum[128];

    // Stage router weights; rows 8..15 are zero padding for the N dimension.
    for (int i = threadIdx.x; i < 16 * ND; i += 256)
        wlds[i >> 10][i & 1023] = (i < NE * ND) ? W[i] : 0.0f;
    __syncthreads();

    const int lane = threadIdx.x & 31;
    const int wv   = threadIdx.x >> 5;
    const int m    = lane & 15;      // token row (A) / column (B,C,D)
    const int kh   = lane >> 4;      // half-select: float4 at col k + 4*kh
    const int row  = blockIdx.x * 128 + wv * 16 + m;
    const float* xrow = x + (size_t)row * ND + 4 * kh;
    const float* wrow = &wlds[m][4 * kh];

    v8f c = {};
    v4f a = *(const v4f*)(xrow);
    v4f b = *(const v4f*)(wrow);
    #pragma unroll 4
    for (int k = 0; k < ND - 8; k += 8) {
        v4f an = *(const v4f*)(xrow + k + 8);   // prefetch next pair
        v4f bn = *(const v4f*)(wrow + k + 8);
        v2f a0 = __builtin_shufflevector(a, a, 0, 1);
        v2f a1 = __builtin_shufflevector(a, a, 2, 3);
        v2f b0 = __builtin_shufflevector(b, b, 0, 1);
        v2f b1 = __builtin_shufflevector(b, b, 2, 3);
        c = __builtin_amdgcn_wmma_f32_16x16x4_f32(false, a0, false, b0,
                                                  (short)0, c, false, false);
        c = __builtin_amdgcn_wmma_f32_16x16x4_f32(false, a1, false, b1,
                                                  (short)0, c, false, false);
        a = an;
        b = bn;
    }
    {
        v2f a0 = __builtin_shufflevector(a, a, 0, 1);
        v2f a1 = __builtin_shufflevector(a, a, 2, 3);
        v2f b0 = __builtin_shufflevector(b, b, 0, 1);
        v2f b1 = __builtin_shufflevector(b, b, 2, 3);
        c = __builtin_amdgcn_wmma_f32_16x16x4_f32(false, a0, false, b0,
                                                  (short)0, c, false, false);
        c = __builtin_amdgcn_wmma_f32_16x16x4_f32(false, a1, false, b1,
                                                  (short)0, c, false, false);
    }

    // Scatter D-matrix to LDS as [token][expert], add bias.
    // C/D layout: lane holds column N = lane&15, rows M = (lane>>4)*8 + i.
    if (m < NE) {
        const float bi = bias[m];
        #pragma unroll
        for (int i = 0; i < 8; ++i)
            lg[wv * 16 + kh * 8 + i][m] = c[i] + bi;
    }
    __syncthreads();

    // Per-token softmax + logsumexp^2 (threads 0..127 -> one token each).
    if (threadIdx.x < 128) {
        const int tt = threadIdx.x;
        const int grow = blockIdx.x * 128 + tt;
        float mx = lg[tt][0];
        #pragma unroll
        for (int e = 1; e < NE; ++e) mx = fmaxf(mx, lg[tt][e]);
        float ex[NE], s = 0.0f;
        #pragma unroll
        for (int e = 0; e < NE; ++e) { ex[e] = expf(lg[tt][e] - mx); s += ex[e]; }
        const float inv = 1.0f / s;
        #pragma unroll
        for (int e = 0; e < NE; ++e) probs[(size_t)grow * NE + e] = ex[e] * inv;
        const float lse = mx + logf(s);
        zsum[tt] = lse * lse;
    }
    __syncthreads();
    if (threadIdx.x == 0) {
        float s = 0.0f;
        for (int i = 0; i < 128; ++i) s += zsum[i];
        zpartial[blockIdx.x] = s;
    }
}

// Fixed-order final reduction -> deterministic z-loss.
__global__ void zfinal_kernel(const float* __restrict__ zpartial,
                              float* __restrict__ out_z) {
    if (threadIdx.x == 0 && blockIdx.x == 0) {
        float s = 0.0f;
        for (int i = 0; i < NBLK1; ++i) s += zpartial[i];
        *out_z = s / (float)NGT;
    }
}

// ---------------------------------------------------------------------------
// Experts-preferred routing. One block per group (1024 threads).
// Phase 1: one concurrent bitonic sort of all 8 expert lists
//          (descending prob, ties -> ascending token index == lax.top_k order).
// Phase 2: sequential over experts: availability prefix-sum gives each still-
//          available token its capacity slot; overflow slots (capacity >
//          remaining tokens) are filled with zero-gate entries in ascending
//          token-index order, exactly matching top_k over masked-to-zero probs.
// ---------------------------------------------------------------------------
__global__ __launch_bounds__(1024) void routing_kernel(
    const float* __restrict__ probs, int cap,
    int* __restrict__ dispatch, float* __restrict__ combine)
{
    __shared__ float svals[NE][NT];   // 64 KiB
    __shared__ short sidx[NE][NT];    // 32 KiB
    __shared__ float avail[NT];       //  8 KiB
    __shared__ int   scan[2][NT];     // 16 KiB (ping-pong prefix sums)
    __shared__ int   pool[NT];        //  8 KiB  -> 128 KiB total (<320 KiB/WG)

    const int g = blockIdx.x;
    const int tid = threadIdx.x;
    const float* gp = probs + (size_t)g * NT * NE;

    for (int t = tid; t < NT; t += 1024) {
        avail[t] = 1.0f;
        #pragma unroll
        for (int e = 0; e < NE; ++e) {
            svals[e][t] = gp[(size_t)t * NE + e];
            sidx[e][t]  = (short)t;
        }
    }
    __syncthreads();

    // Concurrent bitonic sort of the 8 lists.
    for (int k = 2; k <= NT; k <<= 1) {
        for (int j = k >> 1; j > 0; j >>= 1) {
            #pragma unroll
            for (int half = 0; half < 2; ++half) {
                const int i = tid + half * 1024;
                const int ixj = i ^ j;
                if (ixj > i) {
                    const bool up = ((i & k) == 0);
                    #pragma unroll
                    for (int e = 0; e < NE; ++e) {
                        float va = svals[e][i], vb = svals[e][ixj];
                        short ia = sidx[e][i],  ib = sidx[e][ixj];
                        // true if element i ranks AFTER ixj in desired order
                        bool after = (va < vb) || (va == vb && ia > ib);
                        if (after == up) {
                            svals[e][i] = vb; svals[e][ixj] = va;
                            sidx[e][i]  = ib; sidx[e][ixj]  = ia;
                        }
                    }
                }
            }
            __syncthreads();
        }
    }

    for (int e = 0; e < NE; ++e) {
        // Flags (availability in sorted order) + zero-pool (by token index).
        for (int t = tid; t < NT; t += 1024) {
            scan[0][t] = (avail[sidx[e][t]] != 0.0f) ? 1 : 0;
            pool[t]    = (avail[t] == 0.0f) ? 1 : 0;
        }
        __syncthreads();
        // Hillis-Steele inclusive prefix sum over sorted-order flags.
        int src = 0;
        for (int off = 1; off < NT; off <<= 1) {
            for (int t = tid; t < NT; t += 1024) {
                int v = scan[src][t];
                if (t >= off) v += scan[src][t - off];
                scan[1 - src][t] = v;
            }
            __syncthreads();
            src = 1 - src;
        }
        const int A = scan[src][NT - 1];  // # available tokens this round
        __syncthreads();

        // Assign slots to available tokens; clear their availability inline
        // (each token appears exactly once in the sorted list -> race-free).
        for (int t = tid; t < NT; t += 1024) {
            const int tok = sidx[e][t];
            const bool f = (avail[tok] != 0.0f);
            const int r = scan[src][t] - (f ? 1 : 0);  // exclusive rank
            if (f && r < cap) {
                size_t base =
                    ((((size_t)g * NT + tok) * NE) + e) * (size_t)cap + r;
                dispatch[base] = 1;
                combine[base]  = svals[e][t];
                avail[tok] = 0.0f;
            }
        }
        __syncthreads();

        // Overflow: capacity exceeds available tokens -> top_k picks the
        // zero-valued (previously unavailable) tokens, lowest index first.
        if (A < cap) {
            for (int t = tid; t < NT; t += 1024) scan[0][t] = pool[t];
            __syncthreads();
            src = 0;
            for (int off = 1; off < NT; off <<= 1) {
                for (int t = tid; t < NT; t += 1024) {
                    int v = scan[src][t];
                    if (t >= off) v += scan[src][t - off];
                    scan[1 - src][t] = v;
                }
                __syncthreads();
                src = 1 - src;
            }
            for (int t = tid; t < NT; t += 1024) {
                if (pool[t]) {
                    const int r2 = scan[src][t] - 1;  // exclusive rank in pool
                    if (r2 < cap - A) {
                        size_t base =
                            ((((size_t)g * NT + t) * NE) + e) * (size_t)cap +
                            (A + r2);
                        dispatch[base] = 1;
                        combine[base]  = 0.0f;   // gate of a masked-out token
                    }
                }
            }
            __syncthreads();
        }
    }
}

extern "C" void kernel_launch(void* const* d_in, const int* in_sizes, int n_in,
                              void* d_out, int out_size, void* d_ws, size_t ws_size,
                              hipStream_t stream) {
    (void)in_sizes; (void)n_in; (void)ws_size;
    const float* x    = (const float*)d_in[0];
    const float* W    = (const float*)d_in[1];
    const float* bias = (const float*)d_in[2];

    // out_size = 2*N + 3 with N = G*T*E*cap  ->  recover cap on the host.
    const long long N = ((long long)out_size - 3) / 2;
    const int cap = (int)(N / ((long long)NGT * NE));

    float* probs    = (float*)d_ws;                       // [G*T, E]
    float* zpartial = probs + (size_t)NGT * NE;           // [64]

    int*   dispatch = (int*)d_out;                        // [N] int32
    float* combine  = (float*)d_out + N;                  // [N] f32
    float* losses   = (float*)d_out + 2 * N;              // [aux, z, causal]

    zero_kernel<<<512, 256, 0, stream>>>((int*)d_out, (long long)out_size);
    logits_kernel<<<NBLK1, 256, 0, stream>>>(x, W, bias, probs, zpartial);
    zfinal_kernel<<<1, 32, 0, stream>>>(zpartial, losses + 1);
    routing_kernel<<<NG, 1024, 0, stream>>>(probs, cap, dispatch, combine);
}